// LSConv_44830868635853
// MI455X (gfx1250) — compile-verified
//
#include <hip/hip_runtime.h>

// ---- problem constants (match reference) ----
#define Bn   8
#define Cn   128
#define Hdim 128
#define Wdim 128
#define HW   16384      // Hdim*Wdim
#define CMn  64         // C/2
#define C4n  144        // 9 * C/GRP
#define CWn  16         // C/GRP
#define EPSV 1e-5f

typedef float v2f __attribute__((ext_vector_type(2)));
typedef float v8f __attribute__((ext_vector_type(8)));

// ---------------------------------------------------------------------------
// Fused 1x1-conv GEMM via V_WMMA_F32_16X16X4_F32.
//   Y[b, o, p] = sum_c Wm[o,c] * f(Xin[b,c,p])   (+ bias[o])
// f = relu(x*scale + shift) when APPLY_AFFINE (folds the preceding GroupNorm).
// Each wave owns one 16-pixel tile and ALL out-channel tiles -> Xin read once.
// GroupNorm statistics (sum, sumsq) of the raw output accumulated per
// (batch, group) via LDS float atomics then one global atomic per group/block.
// ---------------------------------------------------------------------------
template<int IN_C, int OUT_C, int NGROUPS, int GSIZE, bool APPLY_AFFINE, bool HAS_BIAS>
__global__ __launch_bounds__(256)
void gemm1x1_wmma(const float* __restrict__ Xin,
                  const float* __restrict__ Wm,
                  const float* __restrict__ bias,
                  const float* __restrict__ sIn,
                  const float* __restrict__ tIn,
                  float* __restrict__ Y,
                  float* __restrict__ stats) {
  constexpr int NT = OUT_C / 16;           // out-channel tiles per wave
  __shared__ float lsum[NGROUPS];
  __shared__ float lsq[NGROUPS];
  for (int i = threadIdx.x; i < NGROUPS; i += blockDim.x) { lsum[i] = 0.f; lsq[i] = 0.f; }
  __syncthreads();

  const int wave = threadIdx.x >> 5;
  const int lane = threadIdx.x & 31;
  const int gw   = blockIdx.x * 8 + wave;  // global wave id over (b, pixTile)
  const int b    = gw >> 10;               // 1024 pixel tiles per batch
  const int pix0 = (gw & 1023) << 4;

  const int mRow = lane & 15;              // A-matrix row (out channel in tile)
  const int kOff = (lane >> 4) << 1;       // lanes 16-31 hold K=2,3
  const int nCol = lane & 15;              // B/D column (pixel in tile)

  v8f acc[NT] = {};

  const float* xb = Xin + (size_t)b * IN_C * HW;

  for (int k = 0; k < IN_C; k += 4) {
    float x0 = xb[(size_t)(k + kOff)     * HW + pix0 + nCol];
    float x1 = xb[(size_t)(k + kOff + 1) * HW + pix0 + nCol];
    if constexpr (APPLY_AFFINE) {
      const float sc0 = sIn[b * IN_C + k + kOff];
      const float sh0 = tIn[b * IN_C + k + kOff];
      const float sc1 = sIn[b * IN_C + k + kOff + 1];
      const float sh1 = tIn[b * IN_C + k + kOff + 1];
      x0 = fmaxf(fmaf(x0, sc0, sh0), 0.f);
      x1 = fmaxf(fmaf(x1, sc1, sh1), 0.f);
    }
    v2f bm; bm.x = x0; bm.y = x1;
#pragma unroll
    for (int t = 0; t < NT; t++) {
      v2f a;
      a.x = Wm[(size_t)(t * 16 + mRow) * IN_C + k + kOff];
      a.y = Wm[(size_t)(t * 16 + mRow) * IN_C + k + kOff + 1];
      acc[t] = __builtin_amdgcn_wmma_f32_16x16x4_f32(
          /*neg_a=*/false, a, /*neg_b=*/false, bm,
          /*c_mod=*/(short)0, acc[t], /*reuse_a=*/false, /*reuse_b=*/false);
    }
  }

  float* yb = Y + (size_t)b * OUT_C * HW;
#pragma unroll
  for (int t = 0; t < NT; t++) {
#pragma unroll
    for (int j = 0; j < 8; j++) {
      const int m  = j + ((lane >> 4) << 3);      // D row: lanes16-31 -> M+8
      const int ch = t * 16 + m;
      float v = acc[t][j];
      if constexpr (HAS_BIAS) v += bias[ch];
      yb[(size_t)ch * HW + pix0 + nCol] = v;
      atomicAdd(&lsum[ch / GSIZE], v);
      atomicAdd(&lsq[ch / GSIZE], v * v);
    }
  }
  __syncthreads();
  for (int g = threadIdx.x; g < NGROUPS; g += blockDim.x) {
    atomicAdd(&stats[2 * (b * NGROUPS + g) + 0], lsum[g]);
    atomicAdd(&stats[2 * (b * NGROUPS + g) + 1], lsq[g]);
  }
}

// ---------------------------------------------------------------------------
// stats (sum,sumsq) -> per-(b,c) affine: scale = rsig*gamma, shift = beta-mu*rsig*gamma
// ---------------------------------------------------------------------------
__global__ void finstats(const float* __restrict__ stats,
                         const float* __restrict__ gamma,
                         const float* __restrict__ beta,
                         float* __restrict__ sc, float* __restrict__ sh,
                         int Cc, int gsize, float invN) {
  const int i = blockIdx.x * blockDim.x + threadIdx.x;
  if (i >= Bn * Cc) return;
  const int b = i / Cc, c = i % Cc;
  const int ng = Cc / gsize;
  const int g  = c / gsize;
  const float sum = stats[2 * (b * ng + g) + 0];
  const float sq  = stats[2 * (b * ng + g) + 1];
  const float mu  = sum * invN;
  const float var = sq * invN - mu * mu;
  const float rs  = rsqrtf(var + EPSV);
  const float gm  = gamma[c];
  sc[i] = rs * gm;
  sh[i] = beta[c] - mu * rs * gm;
}

// ---------------------------------------------------------------------------
// Depthwise 7x7, pad 3. Applies GN1 affine + ReLU while staging the halo tile
// in LDS. One block = one (b, c, 16x16 tile). Accumulates GN2 stats.
// ---------------------------------------------------------------------------
#define TS 16
__global__ __launch_bounds__(256)
void dw7x7(const float* __restrict__ Hin, const float* __restrict__ w2,
           const float* __restrict__ s1, const float* __restrict__ t1,
           float* __restrict__ Hout, float* __restrict__ stats) {
  __shared__ float tile[TS + 6][TS + 6 + 1];
  __shared__ float bsum, bsq;
  if (threadIdx.x == 0) { bsum = 0.f; bsq = 0.f; }

  const int blk = blockIdx.x;
  const int tw = blk & 7;
  const int th = (blk >> 3) & 7;
  const int c  = (blk >> 6) & (CMn - 1);
  const int b  = blk >> 12;

  const float* src = Hin + ((size_t)b * CMn + c) * HW;
  const float sc = s1[b * CMn + c];
  const float sh = t1[b * CMn + c];
  const int h0 = th * TS - 3, w0 = tw * TS - 3;

  for (int i = threadIdx.x; i < (TS + 6) * (TS + 6); i += blockDim.x) {
    const int ih = i / (TS + 6), iw = i % (TS + 6);
    const int gh = h0 + ih, gww = w0 + iw;
    float v = 0.f;
    if (gh >= 0 && gh < Hdim && gww >= 0 && gww < Wdim) v = src[gh * Wdim + gww];
    tile[ih][iw] = fmaxf(fmaf(v, sc, sh), 0.f);
  }
  __syncthreads();

  const int ly = threadIdx.x >> 4, lx = threadIdx.x & 15;
  const float* wc = w2 + c * 49;
  float a = 0.f;
#pragma unroll
  for (int ki = 0; ki < 7; ki++)
#pragma unroll
    for (int kj = 0; kj < 7; kj++)
      a = fmaf(tile[ly + ki][lx + kj], wc[ki * 7 + kj], a);

  Hout[((size_t)b * CMn + c) * HW + (th * TS + ly) * Wdim + tw * TS + lx] = a;

  float vs = a, vq = a * a;
  for (int o = 16; o > 0; o >>= 1) { vs += __shfl_xor(vs, o, 32); vq += __shfl_xor(vq, o, 32); }
  if ((threadIdx.x & 31) == 0) { atomicAdd(&bsum, vs); atomicAdd(&bsq, vq); }
  __syncthreads();
  if (threadIdx.x == 0) {
    const int g = c >> 3;                       // GN2: 8 channels/group
    atomicAdd(&stats[2 * (b * 8 + g) + 0], bsum);
    atomicAdd(&stats[2 * (b * 8 + g) + 1], bsq);
  }
}

// ---------------------------------------------------------------------------
// SKA: out[b,c,p] = sum_k x_pad[b,c,h+ki-1,w+kj-1] * (wk[b,(c%16)*9+k,p]*s4+t4)
// One block = 256 consecutive pixels of one (b,c). GN5 stats accumulated.
// wk (75.5 MB) is L2-resident (192 MB) across its 8x channel reuse.
// ---------------------------------------------------------------------------
__global__ __launch_bounds__(256)
void ska(const float* __restrict__ x, const float* __restrict__ wk,
         const float* __restrict__ s4, const float* __restrict__ t4,
         float* __restrict__ out, float* __restrict__ stats) {
  __shared__ float bsum, bsq;
  if (threadIdx.x == 0) { bsum = 0.f; bsq = 0.f; }
  __syncthreads();

  const int blk = blockIdx.x;
  const int pt = blk & 63;
  const int c  = (blk >> 6) & (Cn - 1);
  const int b  = blk >> 13;
  const int p  = pt * 256 + threadIdx.x;
  const int h = p >> 7, w = p & 127;

  const float* xc  = x + ((size_t)b * Cn + c) * HW;
  const float* wkb = wk + (size_t)b * C4n * HW;
  const int cw = c & (CWn - 1);

  float a = 0.f;
#pragma unroll
  for (int ki = 0; ki < 3; ki++) {
    const int hh = h + ki - 1;
#pragma unroll
    for (int kj = 0; kj < 3; kj++) {
      const int ww = w + kj - 1;
      float xv = 0.f;
      if (hh >= 0 && hh < Hdim && ww >= 0 && ww < Wdim) xv = xc[hh * Wdim + ww];
      const int c4 = cw * 9 + ki * 3 + kj;
      const float wv = fmaf(wkb[(size_t)c4 * HW + p], s4[b * C4n + c4], t4[b * C4n + c4]);
      a = fmaf(xv, wv, a);
    }
  }
  out[((size_t)b * Cn + c) * HW + p] = a;

  float vs = a, vq = a * a;
  for (int o = 16; o > 0; o >>= 1) { vs += __shfl_xor(vs, o, 32); vq += __shfl_xor(vq, o, 32); }
  if ((threadIdx.x & 31) == 0) { atomicAdd(&bsum, vs); atomicAdd(&bsq, vq); }
  __syncthreads();
  if (threadIdx.x == 0) {
    const int g = c >> 4;                        // GN5: 16 channels/group
    atomicAdd(&stats[2 * (b * 8 + g) + 0], bsum);
    atomicAdd(&stats[2 * (b * 8 + g) + 1], bsq);
  }
}

// out = out*scale5 + shift5 + x  (in place, final GN + residual)
__global__ __launch_bounds__(256)
void finalgn(const float* __restrict__ x, const float* __restrict__ s5,
             const float* __restrict__ t5, float* __restrict__ out) {
  const size_t i = (size_t)blockIdx.x * blockDim.x + threadIdx.x;
  const int bc = (int)(i / HW);
  out[i] = fmaf(out[i], s5[bc], t5[bc]) + x[i];
}

// ---------------------------------------------------------------------------
extern "C" void kernel_launch(void* const* d_in, const int* in_sizes, int n_in,
                              void* d_out, int out_size, void* d_ws, size_t ws_size,
                              hipStream_t stream) {
  (void)in_sizes; (void)n_in; (void)out_size; (void)ws_size;
  const float* x    = (const float*)d_in[0];
  const float* w1   = (const float*)d_in[1];
  const float* g1   = (const float*)d_in[2];
  const float* b1   = (const float*)d_in[3];
  const float* w2   = (const float*)d_in[4];
  const float* g2   = (const float*)d_in[5];
  const float* b2   = (const float*)d_in[6];
  const float* w3   = (const float*)d_in[7];
  const float* g3   = (const float*)d_in[8];
  const float* b3   = (const float*)d_in[9];
  const float* w4   = (const float*)d_in[10];
  const float* bias4= (const float*)d_in[11];
  const float* g4   = (const float*)d_in[12];
  const float* b4   = (const float*)d_in[13];
  const float* gng  = (const float*)d_in[14];
  const float* gnb  = (const float*)d_in[15];
  float* out = (float*)d_out;
  float* ws  = (float*)d_ws;

  const size_t A = (size_t)Bn * CMn * HW;        // 8388608 floats
  float* h1  = ws;                               // also reused as h3
  float* h2  = ws + A;
  float* wkb = ws + 2 * A;                       // B*144*HW = 18874368 floats
  float* st  = ws + 2 * A + (size_t)Bn * C4n * HW;
  float *st1 = st, *st2 = st + 256, *st3 = st + 512, *st4 = st + 768, *st5 = st + 1024;
  float* scb = st + 1280;
  float *s1 = scb,        *t1 = scb + 512,
        *s2 = scb + 1024, *t2 = scb + 1536,
        *s3 = scb + 2048, *t3 = scb + 2560,
        *s4 = scb + 3072, *t4 = scb + 4224,   // 1152 each
        *s5 = scb + 5376, *t5 = scb + 6400;   // 1024 each

  hipMemsetAsync(st, 0, 1280 * sizeof(float), stream);

  // h1 = w1 . x  (+GN1 stats)
  gemm1x1_wmma<Cn, CMn, 8, 8, false, false><<<1024, 256, 0, stream>>>(
      x, w1, nullptr, nullptr, nullptr, h1, st1);
  finstats<<<2, 256, 0, stream>>>(st1, g1, b1, s1, t1, CMn, 8, 1.0f / (8.0f * HW));

  // h2 = dw7x7(relu(affine1(h1)))  (+GN2 stats)
  dw7x7<<<Bn * CMn * 64, 256, 0, stream>>>(h1, w2, s1, t1, h2, st2);
  finstats<<<2, 256, 0, stream>>>(st2, g2, b2, s2, t2, CMn, 8, 1.0f / (8.0f * HW));

  // h3 (=h1 buf) = w3 . relu(affine2(h2))  (+GN3 stats)
  gemm1x1_wmma<CMn, CMn, 8, 8, true, false><<<1024, 256, 0, stream>>>(
      h2, w3, nullptr, s2, t2, h1, st3);
  finstats<<<2, 256, 0, stream>>>(st3, g3, b3, s3, t3, CMn, 8, 1.0f / (8.0f * HW));

  // wk = w4 . relu(affine3(h3)) + bias4  (+GN4 stats, groups of 9)
  gemm1x1_wmma<CMn, C4n, 16, 9, true, true><<<1024, 256, 0, stream>>>(
      h1, w4, bias4, s3, t3, wkb, st4);
  finstats<<<5, 256, 0, stream>>>(st4, g4, b4, s4, t4, C4n, 9, 1.0f / (9.0f * HW));

  // out_raw = SKA(x, affine4(wk))  (+GN5 stats)
  ska<<<Bn * Cn * 64, 256, 0, stream>>>(x, wkb, s4, t4, out, st5);
  finstats<<<4, 256, 0, stream>>>(st5, gng, gnb, s5, t5, Cn, 16, 1.0f / (16.0f * HW));

  // out = affine5(out_raw) + x
  finalgn<<<65536, 256, 0, stream>>>(x, s5, t5, out);
}